// MaxMinPool_70145405878420
// MI455X (gfx1250) — compile-verified
//
#include <hip/hip_runtime.h>
#include <cstdint>

// Tropical (max-plus / min-plus) matmul for MI455X (gfx1250).
//   x: (2048, 512) f32, w: (512, 256) f32, out: (2048, 256) f32
//   out[b,u] = max_f (x[b,f]+w[f,u])  for u < 128
//   out[b,u] = min_f (x[b,f]+w[f,u])  for u >= 128
// Compute-bound on VALU (add + max/min); WMMA cannot express the tropical
// semiring. CDNA5-specific paths: async global->LDS copy engine (ASYNCcnt,
// s_wait_asynccnt), split barrier signal/wait, VOPD dual-issue, v_max3/v_min3
// 3-input reductions (two k-steps folded per accumulator update).
// Single compute-tile instance (no tail peel) to keep static code compact.

#define BATCH 2048
#define KDIM  512
#define NDIM  256
#define BM    32          // rows per block tile
#define BN    64          // cols per block tile
#define BK    16          // K per tile
#define NT    (KDIM / BK) // 32 K-tiles
#define XS    36          // padded LDS stride for x tile (floats), 36*4 % 16 == 0
#define WS    68          // padded LDS stride for w tile (floats), 68*4 % 16 == 0
#define XTILE (BK * XS)   // 576 floats per x buffer
#define WTILE (BK * WS)   // 1088 floats per w buffer
#define LDS_FLOATS (2 * XTILE + 2 * WTILE) // 3328 floats = 13312 B

__device__ __forceinline__ void async_ld_b32(uint32_t lds_addr, uint32_t voff,
                                             uint64_t sbase) {
  asm volatile("global_load_async_to_lds_b32 %0, %1, %2"
               :: "v"(lds_addr), "v"(voff), "s"(sbase) : "memory");
}
__device__ __forceinline__ void async_ld_b128(uint32_t lds_addr, uint32_t voff,
                                              uint64_t sbase) {
  asm volatile("global_load_async_to_lds_b128 %0, %1, %2"
               :: "v"(lds_addr), "v"(voff), "s"(sbase) : "memory");
}
__device__ __forceinline__ void wait_async_zero() {
  asm volatile("s_wait_asynccnt 0x0" ::: "memory");
}

template <bool IS_MAX>
__device__ __forceinline__ void tropical_body(const float* __restrict__ x,
                                              const float* __restrict__ w,
                                              float* __restrict__ out,
                                              float* lds) {
  const int tid = threadIdx.x;          // 0..127
  const int tx  = tid & 15;             // 16 threads across N
  const int ty  = tid >> 4;             // 8 threads across M
  const int row0 = blockIdx.x * BM;     // 0..2016
  const int col0 = blockIdx.y * BN;     // 0,64,128,192

  // LDS byte base (low 32 bits of the generic LDS pointer == LDS byte offset)
  const uint32_t lds_base = (uint32_t)(uintptr_t)lds;
  const uint32_t xbuf_b[2] = { lds_base,
                               lds_base + (uint32_t)XTILE * 4u };
  const uint32_t wbuf_b[2] = { lds_base + (uint32_t)(2 * XTILE) * 4u,
                               lds_base + (uint32_t)(2 * XTILE + WTILE) * 4u };
  const uint64_t xg = (uint64_t)(uintptr_t)x;
  const uint64_t wg = (uint64_t)(uintptr_t)w;

  // Staging thread mapping
  const int sk  = tid & 15;       // x: k within tile (0..15)
  const int sm  = tid >> 4;       // x: m base (0..7), covers m = sm + 8*j
  const int wn4 = (tid & 15) * 4; // w: 4-col group
  const int wk  = tid >> 4;       // w: k row base (0..7), covers kr = wk + 8*jj

  auto stage = [&](int kt, int buf) {
    const int k0 = kt * BK;
    // w tile: lds_w[k][n], two b128 per thread, global rows are contiguous
#pragma unroll
    for (int jj = 0; jj < 2; ++jj) {
      const int kr = wk + 8 * jj;
      uint32_t wl = wbuf_b[buf] + (uint32_t)(kr * WS + wn4) * 4u;
      uint32_t wo = (uint32_t)(((k0 + kr) * NDIM + col0 + wn4) * 4);
      async_ld_b128(wl, wo, wg);
    }
    // x tile transposed on the fly: lds_x[k][m], four b32 per thread
#pragma unroll
    for (int j = 0; j < 4; ++j) {
      const int m = sm + 8 * j;
      uint32_t xl = xbuf_b[buf] + (uint32_t)(sk * XS + m) * 4u;
      uint32_t xo = (uint32_t)(((row0 + m) * KDIM + k0 + sk) * 4);
      async_ld_b32(xl, xo, xg);
    }
  };

  float acc[4][4];
#pragma unroll
  for (int i = 0; i < 4; ++i)
#pragma unroll
    for (int j = 0; j < 4; ++j)
      acc[i][j] = IS_MAX ? -__builtin_inff() : __builtin_inff();

  stage(0, 0);
  wait_async_zero();
  __syncthreads();

#pragma unroll 1
  for (int kt = 0; kt < NT; ++kt) {
    const int buf = kt & 1;
    if (kt + 1 < NT) stage(kt + 1, buf ^ 1);   // prefetch next tile (async)

    const float* xs = &lds[buf * XTILE + 4 * ty];
    const float* ws = &lds[2 * XTILE + buf * WTILE + 4 * tx];
    // Two k-steps folded per accumulator update -> v_max3/v_min3 eligible.
#pragma unroll
    for (int k = 0; k < BK; k += 2) {
      const float4 xv0 = *reinterpret_cast<const float4*>(xs + k * XS);
      const float4 xv1 = *reinterpret_cast<const float4*>(xs + (k + 1) * XS);
      const float4 wv0 = *reinterpret_cast<const float4*>(ws + k * WS);
      const float4 wv1 = *reinterpret_cast<const float4*>(ws + (k + 1) * WS);
      const float xa0[4] = { xv0.x, xv0.y, xv0.z, xv0.w };
      const float xa1[4] = { xv1.x, xv1.y, xv1.z, xv1.w };
      const float wb0[4] = { wv0.x, wv0.y, wv0.z, wv0.w };
      const float wb1[4] = { wv1.x, wv1.y, wv1.z, wv1.w };
#pragma unroll
      for (int i = 0; i < 4; ++i)
#pragma unroll
        for (int j = 0; j < 4; ++j) {
          const float s0 = xa0[i] + wb0[j];
          const float s1 = xa1[i] + wb1[j];
          if (IS_MAX) acc[i][j] = fmaxf(acc[i][j], fmaxf(s0, s1)); // v_max3
          else        acc[i][j] = fminf(acc[i][j], fminf(s0, s1)); // v_min3
        }
    }

    wait_async_zero();   // tile kt+1 resident before next iteration reads it
    __syncthreads();     // all waves done reading buf before it is overwritten
  }

#pragma unroll
  for (int i = 0; i < 4; ++i) {
    const int r = row0 + 4 * ty + i;
    float4 o;
    o.x = acc[i][0]; o.y = acc[i][1]; o.z = acc[i][2]; o.w = acc[i][3];
    *reinterpret_cast<float4*>(&out[r * NDIM + col0 + 4 * tx]) = o;
  }
}

__global__ __launch_bounds__(128)
void MaxMinPool_70145405878420_kernel(const float* __restrict__ x,
                                      const float* __restrict__ w,
                                      float* __restrict__ out) {
  __shared__ __align__(16) float lds[LDS_FLOATS];
  // blockIdx.y in {0,1} -> max-plus half (cols 0..127)
  // blockIdx.y in {2,3} -> min-plus half (cols 128..255)
  if (blockIdx.y < 2) tropical_body<true >(x, w, out, lds);
  else                tropical_body<false>(x, w, out, lds);
}

extern "C" void kernel_launch(void* const* d_in, const int* in_sizes, int n_in,
                              void* d_out, int out_size, void* d_ws, size_t ws_size,
                              hipStream_t stream) {
  (void)in_sizes; (void)n_in; (void)d_ws; (void)ws_size; (void)out_size;
  const float* x = (const float*)d_in[0];   // (2048, 512)
  const float* w = (const float*)d_in[1];   // (512, 256)
  float* out     = (float*)d_out;           // (2048, 256)

  dim3 block(128);
  dim3 grid(BATCH / BM, NDIM / BN);         // (64, 4) = 256 workgroups
  MaxMinPool_70145405878420_kernel<<<grid, block, 0, stream>>>(x, w, out);
}